// ArbitraryBatchTimeSeriesInterpolator_1322849927844
// MI455X (gfx1250) — compile-verified
//
#include <hip/hip_runtime.h>
#include <stdint.h>

// Problem geometry (matches reference).
#define NTIME   4096
#define NBATCH  4096
#define NEXTRA  256

// Tiling.
#define COLS    128                 // batch columns per block (== blockDim.x)
#define CSTRIDE 128                 // coarse sampling stride along time axis
#define NCOARSE (NTIME / CSTRIDE)   // 32 coarse samples per column
#define QPB     8                   // queries per block

// out[q,b] = v0 + slope * (t[q,b] - t0), where
//   counts = searchsorted_right(times[:,b], t[q,b])
//   gi = counts % NTIME; idx = (gi-1) mod NTIME; sidx = (gi-1) mod (NTIME-1)
//
// Hierarchical search:
//   level 0: 32 coarse samples (every 128th time) staged in LDS via CDNA5
//            async global->LDS copies (ASYNCcnt / s_wait_asynccnt),
//            searched with 5 fixed branchless steps (ds_load + v_cndmask).
//   level 1: 7 fixed branchless steps over a 128-row global window.
// Query loop unrolled 4x -> 4 independent probe chains per thread for MLP.
__global__ __launch_bounds__(COLS)
void ts_interp_kernel(const float* __restrict__ times,
                      const float* __restrict__ values,
                      const float* __restrict__ tq,
                      float* __restrict__ out)
{
    __shared__ float coarse[NCOARSE * COLS];   // 16 KB

    const int tid   = threadIdx.x;
    const int cBase = blockIdx.x * COLS;
    const int qBase = blockIdx.y * QPB;
    const int c     = cBase + tid;             // this thread's batch column

    // ---- Stage coarse samples times[r*CSTRIDE, c] into LDS -------------
    // CDNA5 async copy path: tracked by ASYNCcnt, completed with
    // s_wait_asynccnt, made visible to all waves by the barrier.
    {
        #pragma unroll
        for (int r = 0; r < NCOARSE; ++r) {
            // Low 32 bits of the generic pointer == wave-relative LDS offset.
            unsigned ldsAddr = (unsigned)(uintptr_t)(&coarse[r * COLS + tid]);
            unsigned long long gAddr =
                (unsigned long long)(uintptr_t)(times + (size_t)(r * CSTRIDE) * NBATCH + c);
            asm volatile("global_load_async_to_lds_b32 %0, %1, off"
                         :: "v"(ldsAddr), "v"(gAddr)
                         : "memory");
        }
        asm volatile("s_wait_asynccnt 0x0" ::: "memory");
    }
    __syncthreads();

    #pragma unroll 4
    for (int qi = 0; qi < QPB; ++qi) {
        const int q = qBase + qi;
        const float t = tq[(size_t)q * NBATCH + c];

        // ---- Coarse search in LDS (branchless, 5 fixed steps) ----------
        // k = #{r in [0,31): coarse[r] <= t}  (array length 32 = 2^5)
        // LDS index = m*COLS + tid -> bank = tid % 64: conflict-free.
        int k = 0;
        #pragma unroll
        for (int step = NCOARSE / 2; step >= 1; step >>= 1) {
            float v = coarse[(k + step - 1) * COLS + tid];
            k = (v <= t) ? (k + step) : k;
        }
        // k in [0, NCOARSE]

        // ---- Fine search in global (branchless, 7 fixed steps) ---------
        // Window is exactly CSTRIDE rows for every k >= 1 (incl. k == 32).
        // For k == 0 the result is forced to 0 below; probes stay in-bounds.
        const int base = (k == 0) ? 0 : (k - 1) * CSTRIDE;
        int ofs = 0;
        #pragma unroll
        for (int step = CSTRIDE / 2; step >= 1; step >>= 1) {
            float v = times[(size_t)(base + ofs + step - 1) * NBATCH + c];
            ofs = (v <= t) ? (ofs + step) : ofs;
        }
        const int counts = (k == 0) ? 0 : (base + ofs);   // in [0, NTIME]

        // ---- Index arithmetic with torch-style negative wrap -----------
        const int gi   = counts & (NTIME - 1);            // counts % NTIME (pow2)
        const int sidx = (gi == 0) ? (NTIME - 2) : (gi - 1);  // (gi-1) mod (NTIME-1)
        // idx = (gi-1) mod NTIME:  gi==0 -> NTIME-1 == sidx+1; else gi-1 == sidx.

        // ---- Gather the two slope endpoints (covers idx too) -----------
        const float ta = times [(size_t)sidx       * NBATCH + c];
        const float tb = times [(size_t)(sidx + 1) * NBATCH + c];
        const float va = values[(size_t)sidx       * NBATCH + c];
        const float vb = values[(size_t)(sidx + 1) * NBATCH + c];

        const float t0 = (gi == 0) ? tb : ta;        // times[idx]
        const float v0 = (gi == 0) ? vb : va;        // values[idx]
        const float s  = (vb - va) / (tb - ta);      // slopes[sidx]

        out[(size_t)q * NBATCH + c] = v0 + s * (t - t0);
    }
}

extern "C" void kernel_launch(void* const* d_in, const int* in_sizes, int n_in,
                              void* d_out, int out_size, void* d_ws, size_t ws_size,
                              hipStream_t stream) {
    (void)in_sizes; (void)n_in; (void)out_size; (void)d_ws; (void)ws_size;
    const float* times  = (const float*)d_in[0];
    const float* values = (const float*)d_in[1];
    const float* t      = (const float*)d_in[2];
    float* out          = (float*)d_out;

    dim3 grid(NBATCH / COLS, NEXTRA / QPB);   // (32, 32) = 1024 blocks
    dim3 block(COLS);                          // 128 threads = 4 wave32
    ts_interp_kernel<<<grid, block, 0, stream>>>(times, values, t, out);
}